// Embeddings_21792664060221
// MI455X (gfx1250) — compile-verified
//
#include <hip/hip_runtime.h>

typedef __attribute__((ext_vector_type(2))) float v2f;
typedef __attribute__((ext_vector_type(8))) float v8f;

#define NEG_SLOPE 0.2f

// float atomic max via int/uint monotone mapping (works for mixed signs,
// init value must be -inf = 0xff800000)
__device__ __forceinline__ void atomic_max_f32(float* addr, float v) {
  if (v >= 0.0f) atomicMax((int*)addr, __float_as_int(v));
  else           atomicMin((unsigned int*)addr, (unsigned int)__float_as_int(v));
}

// ---------------- K0: init workspace accumulators ----------------
__global__ void gat_init(float* __restrict__ attr_sum, float* __restrict__ cnt,
                         float* __restrict__ mx, float* __restrict__ den,
                         float* __restrict__ num, int N) {
  int i = blockIdx.x * blockDim.x + threadIdx.x;
  if (i < N) { attr_sum[i] = 0.0f; cnt[i] = 0.0f; }
  if (i < 2 * N) {
    mx[i]  = __int_as_float(0xff800000);  // -inf
    den[i] = 0.0f;
    num[i] = 0.0f;
  }
}

// ---------------- K1: in-degree + incoming-attr sum (for mean self-loop attr)
__global__ void gat_degree(const int* __restrict__ ei,    // [2*E]: src | dst
                           const float* __restrict__ attr,
                           float* __restrict__ attr_sum, float* __restrict__ cnt,
                           int E) {
  int e = blockIdx.x * blockDim.x + threadIdx.x;
  if (e >= E) return;
  int d = ei[E + e];
  atomicAdd(&attr_sum[d], attr[e]);
  atomicAdd(&cnt[d], 1.0f);
}

// ---------------- K2: GATv2 logits via V_WMMA_F32_16X16X4_F32 -------------
// One wave = 16 virtual edges (real edges j<E, self-loops j>=E).
// A (16x4): row = [x[src], x[dst], attr, 1]
// B (4x128, 8 tiles of 4x16): rows = [W_l; W_r; W_e; b_l+b_r]
// G = A@B is the GATv2 pre-activation; then leakyrelu, dot with att per head,
// cross-lane reduce, store logits and atomic segment-max.
__global__ void gat_logits_wmma(
    const float* __restrict__ x,
    const int*   __restrict__ ei,
    const float* __restrict__ attr,
    const float* __restrict__ W_l, const float* __restrict__ b_l,
    const float* __restrict__ W_r, const float* __restrict__ b_r,
    const float* __restrict__ W_e, const float* __restrict__ att,
    const float* __restrict__ attr_sum, const float* __restrict__ cnt,
    float* __restrict__ mx,       // [N*2], pre-init -inf
    float* __restrict__ logits,   // [(E+N)*2]
    int E, int M /* = E + N */) {
  const int wave = (blockIdx.x * blockDim.x + threadIdx.x) >> 5;
  const int lane = threadIdx.x & 31;
  const int base = wave * 16;
  if (base >= M) return;            // wave-uniform exit (EXEC all-1 at WMMA)
  const int  lh = lane & 15;
  const bool hi = lane >= 16;

  // --- B tiles: 4x16 f32. lanes 0-15: {K0,K1} = {W_l,W_r}; lanes 16-31:
  // {K2,K3} = {W_e, b_l+b_r}. Column N = lane%16 -> channel 16*t + lh.
  v2f B[8];
#pragma unroll
  for (int t = 0; t < 8; ++t) {
    int ch = t * 16 + lh;
    if (!hi) { B[t].x = W_l[ch]; B[t].y = W_r[ch]; }
    else     { B[t].x = W_e[ch]; B[t].y = b_l[ch] + b_r[ch]; }
  }

  // --- A: 16x4 f32. lanes 0-15: {K0,K1} = {x_src, x_dst}; lanes 16-31:
  // {K2,K3} = {attr, 1}. Row M = lane%16 -> edge base+lh.
  int j = base + lh;
  float a0 = 0.0f, a1 = 0.0f;
  if (j < M) {
    if (!hi) {
      int s, d;
      if (j < E) { s = ei[j]; d = ei[E + j]; }
      else       { s = d = j - E; }
      a0 = x[s];
      a1 = x[d];
    } else {
      if (j < E) a0 = attr[j];
      else { int n = j - E; a0 = attr_sum[n] / fmaxf(cnt[n], 1.0f); }
      a1 = 1.0f;
    }
  }
  v2f A; A.x = a0; A.y = a1;

  // --- 8 WMMAs + leakyrelu + att weighting. p[h][r]: partial logit for the
  // edge in D-row r of this lane's half, head h.
  float p[2][8];
#pragma unroll
  for (int h = 0; h < 2; ++h)
#pragma unroll
    for (int r = 0; r < 8; ++r) p[h][r] = 0.0f;

#pragma unroll
  for (int t = 0; t < 8; ++t) {
    v8f cz = {};
    // (neg_a, A, neg_b, B, c_mod, C, reuse_a, reuse_b)
    v8f D = __builtin_amdgcn_wmma_f32_16x16x4_f32(
        false, A, false, B[t], (short)0, cz, false, false);
    float av = att[t * 16 + lh];   // channel index == h*64 + c flat
    const int h = t >> 2;          // tiles 0-3 -> head 0, 4-7 -> head 1
#pragma unroll
    for (int r = 0; r < 8; ++r) {
      float g = D[r];
      g = (g > 0.0f) ? g : NEG_SLOPE * g;
      p[h][r] = fmaf(g, av, p[h][r]);
    }
  }

  // --- reduce over the 16 channels of each half (lanes 0-15 hold edges 0-7,
  // lanes 16-31 hold edges 8-15). All lanes participate (no divergence).
#pragma unroll
  for (int off = 1; off < 16; off <<= 1) {
#pragma unroll
    for (int r = 0; r < 8; ++r) {
      p[0][r] += __shfl_xor(p[0][r], off, 32);
      p[1][r] += __shfl_xor(p[1][r], off, 32);
    }
  }

  // --- lanes with lh<8 own one edge each: lower half -> edges 0-7,
  // upper half -> edges 8-15. Select p[*][lh] with cndmask chain.
  float l0 = p[0][0], l1 = p[1][0];
#pragma unroll
  for (int q = 1; q < 8; ++q) {
    bool sel = (lh == q);
    l0 = sel ? p[0][q] : l0;
    l1 = sel ? p[1][q] : l1;
  }
  if (lh < 8) {
    int jm = base + (hi ? 8 + lh : lh);
    if (jm < M) {
      int d = (jm < E) ? ei[E + jm] : (jm - E);
      logits[2 * jm + 0] = l0;
      logits[2 * jm + 1] = l1;
      atomic_max_f32(&mx[2 * d + 0], l0);
      atomic_max_f32(&mx[2 * d + 1], l1);
    }
  }
}

// ---------------- K3: segment softmax accumulation --------------------
// num[n,h] = sum_j exp(logit-mx)*x[src_j];  den[n,h] = sum_j exp(logit-mx)
__global__ void gat_expsum(const float* __restrict__ x,
                           const int* __restrict__ ei,
                           const float* __restrict__ mx,
                           const float* __restrict__ logits,
                           float* __restrict__ den, float* __restrict__ num,
                           int E, int M) {
  int j = blockIdx.x * blockDim.x + threadIdx.x;
  if (j >= M) return;
  int s, d;
  if (j < E) { s = ei[j]; d = ei[E + j]; }
  else       { s = d = j - E; }
  float xs = x[s];
  float a0 = expf(logits[2 * j + 0] - mx[2 * d + 0]);
  float a1 = expf(logits[2 * j + 1] - mx[2 * d + 1]);
  atomicAdd(&den[2 * d + 0], a0);
  atomicAdd(&den[2 * d + 1], a1);
  atomicAdd(&num[2 * d + 0], a0 * xs);
  atomicAdd(&num[2 * d + 1], a1 * xs);
}

// ---------------- K4: rank-1 output reconstruction --------------------
// out[n,k] = (num/den)[n,h(k)] * W_l[k] + b_l[k] + bias[k]   (sum alpha == 1)
__global__ void gat_out(const float* __restrict__ num, const float* __restrict__ den,
                        const float* __restrict__ W_l, const float* __restrict__ b_l,
                        const float* __restrict__ bias, float* __restrict__ out,
                        int N) {
  int i = blockIdx.x * blockDim.x + threadIdx.x;
  if (i >= N * 128) return;
  int k = i & 127;
  int n = i >> 7;
  int h = k >> 6;
  float S = num[2 * n + h] / den[2 * n + h];
  out[i] = fmaf(S, W_l[k], b_l[k] + bias[k]);
}

extern "C" void kernel_launch(void* const* d_in, const int* in_sizes, int n_in,
                              void* d_out, int out_size, void* d_ws, size_t ws_size,
                              hipStream_t stream) {
  const float* x    = (const float*)d_in[0];
  const int*   ei   = (const int*)  d_in[1];   // [2,E] flat: src row then dst row
  const float* attr = (const float*)d_in[2];
  const float* W_l  = (const float*)d_in[3];
  const float* b_l  = (const float*)d_in[4];
  const float* W_r  = (const float*)d_in[5];
  const float* b_r  = (const float*)d_in[6];
  const float* W_e  = (const float*)d_in[7];
  const float* att  = (const float*)d_in[8];
  const float* bias = (const float*)d_in[9];

  const int N = in_sizes[0];      // x is [N,1]
  const int E = in_sizes[2];      // edge_attr is [E,1]
  const int M = E + N;            // edges + self-loops

  // workspace layout (floats): attr_sum[N] cnt[N] mx[2N] den[2N] num[2N] logits[2M]
  float* ws       = (float*)d_ws;
  float* attr_sum = ws;
  float* cnt      = attr_sum + N;
  float* mx       = cnt + N;
  float* den      = mx + 2 * N;
  float* num      = den + 2 * N;
  float* logits   = num + 2 * N;

  const int BLK = 256;

  gat_init<<<(2 * N + BLK - 1) / BLK, BLK, 0, stream>>>(attr_sum, cnt, mx, den, num, N);

  gat_degree<<<(E + BLK - 1) / BLK, BLK, 0, stream>>>(ei, attr, attr_sum, cnt, E);

  // one wave per 16 virtual edges; 8 waves per 256-thread block
  int nGroups = (M + 15) / 16;
  int blocks2 = (nGroups + 7) / 8;
  gat_logits_wmma<<<blocks2, BLK, 0, stream>>>(x, ei, attr, W_l, b_l, W_r, b_r,
                                               W_e, att, attr_sum, cnt, mx,
                                               logits, E, M);

  gat_expsum<<<(M + BLK - 1) / BLK, BLK, 0, stream>>>(x, ei, mx, logits, den, num, E, M);

  gat_out<<<(N * 128 + BLK - 1) / BLK, BLK, 0, stream>>>(num, den, W_l, b_l, bias,
                                                         (float*)d_out, N);
}